// BaselineJKNet_14697378087200
// MI455X (gfx1250) — compile-verified
//
#include <hip/hip_runtime.h>
#include <hip/hip_bf16.h>

typedef __attribute__((ext_vector_type(2))) float v2f;
typedef __attribute__((ext_vector_type(8))) float v8f;

// ---------------------------------------------------------------------------
// WMMA f32 GEMM tile: C[16 x 16] = A[16 x K] @ B[K x LDB] (+bias), one tile
// per wave. All strides are compile-time so every A/B/C access is
// base-pointer + immediate offset (no per-iteration address math).
//
// A-frag (16x4 f32, ISA 7.12.2): lanes 0-15 hold M=lane, K=k+0/k+1;
// lanes 16-31 hold M=lane-16, K=k+2/k+3  -> one b64 load at Arow[k + 2*hi].
// B-frag mirrors on K: lane holds B[k+2*hi][col], B[k+2*hi+1][col].
// C/D: VGPR r -> (M = r + 8*hi, N = lane&15).
// ---------------------------------------------------------------------------
template <int K, int LDA, int LDB, bool COLGUARD>
__global__ void gemm_wmma_k(const float* __restrict__ A,
                            const float* __restrict__ B,
                            const float* __restrict__ bias,
                            float* __restrict__ C, int M) {
  const int lane  = threadIdx.x & 31;
  const int laneM = lane & 15;
  const int hi    = lane >> 4;            // 0 or 1
  const int row0  = blockIdx.x * 16;
  const int col0  = threadIdx.y * 16;     // blockDim.y waves cover the N tiles

  // Clamp row so A loads are unconditional (M is a multiple of 16 in practice).
  int arow = row0 + laneM;
  if (arow > M - 1) arow = M - 1;
  const float* __restrict__ aptr = A + (size_t)arow * LDA + 2 * hi;

  // Column clamp + multiplicative mask instead of predicated loads.
  const int col   = col0 + laneM;
  int       colc  = col;
  float     cmask = 1.0f;
  if (COLGUARD) {
    if (colc > LDB - 1) { colc = LDB - 1; cmask = 0.0f; }
  }
  const float* __restrict__ bptr = B + 2 * hi * LDB + colc;

  v8f acc = {};
#pragma unroll
  for (int k = 0; k < K; k += 4) {
    v2f a = *(const v2f*)(aptr + k);      // global_load_b64, imm offset
    v2f b;
    b.x = bptr[k * LDB];                  // global_load_b32, imm offset
    b.y = bptr[k * LDB + LDB];
    if (COLGUARD) { b.x *= cmask; b.y *= cmask; }
    acc = __builtin_amdgcn_wmma_f32_16x16x4_f32(false, a, false, b,
                                                (short)0, acc, false, false);
  }

  if (COLGUARD && cmask == 0.0f) return;
  const float bc = COLGUARD ? bias[col] : 0.0f;
  float* __restrict__ cptr = C + (size_t)(row0 + 8 * hi) * LDB + col;
#pragma unroll
  for (int r = 0; r < 8; ++r) {
    const int row = row0 + r + 8 * hi;
    if (row < M) cptr[r * LDB] = acc[r] + bc;
  }
}

// ---------------------------------------------------------------------------
// Edge scatter: one wave per edge; lane owns channels {2*lane, 2*lane+1}.
// One b64 gather of T[src], scale by edge weight, two f32 atomic adds.
// ---------------------------------------------------------------------------
__global__ void scatter_edges(const float* __restrict__ T,
                              const int* __restrict__ src,
                              const int* __restrict__ dst,
                              const float* __restrict__ w,
                              float* __restrict__ agg, int E) {
  const int e = blockIdx.x * (blockDim.x >> 5) + (threadIdx.x >> 5);
  if (e >= E) return;
  const int   lane = threadIdx.x & 31;
  const int   s  = src[e];
  const int   d  = dst[e];
  const float we = w[e];
  const v2f   tv = *(const v2f*)(T + (size_t)s * 64 + 2 * lane);
  float* __restrict__ ap = agg + (size_t)d * 64 + 2 * lane;
  atomicAdd(ap,     tv.x * we);
  atomicAdd(ap + 1, tv.y * we);
}

__global__ void zero_f32(float* __restrict__ p, int n) {
  const int i = blockIdx.x * blockDim.x + threadIdx.x;
  if (i < n) p[i] = 0.0f;
}

// agg + bias, ReLU, write into JK concat buffer [N x 192] at column offset.
__global__ void bias_relu_to_jk(const float* __restrict__ agg,
                                const float* __restrict__ b,
                                float* __restrict__ hjk,
                                int n64, int colOff) {
  const int i = blockIdx.x * blockDim.x + threadIdx.x;
  if (i >= n64) return;
  const int row = i >> 6;
  const int c   = i & 63;
  const float v = agg[i] + b[c];
  hjk[(size_t)row * 192 + colOff + c] = fmaxf(v, 0.0f);
}

extern "C" void kernel_launch(void* const* d_in, const int* in_sizes, int n_in,
                              void* d_out, int out_size, void* d_ws, size_t ws_size,
                              hipStream_t stream) {
  const float* x    = (const float*)d_in[0];
  const int*   ei   = (const int*)d_in[1];   // [2, E] : src row then dst row
  const float* ew   = (const float*)d_in[2];
  const float* W1   = (const float*)d_in[3];
  const float* b1   = (const float*)d_in[4];
  const float* W2   = (const float*)d_in[5];
  const float* b2   = (const float*)d_in[6];
  const float* W3   = (const float*)d_in[7];
  const float* b3   = (const float*)d_in[8];
  const float* Wlin = (const float*)d_in[9];
  const float* blin = (const float*)d_in[10];

  const int N = in_sizes[0] / 128;
  const int E = in_sizes[2];
  const int* src = ei;
  const int* dst = ei + E;

  float* T   = (float*)d_ws;              // [N x 64]
  float* agg = T   + (size_t)N * 64;      // [N x 64]
  float* hjk = agg + (size_t)N * 64;      // [N x 192]

  const int Mtiles  = (N + 15) / 16;
  const int n64     = N * 64;
  const int zblocks = (n64 + 255) / 256;
  const int sblocks = (E + 7) / 8;        // 8 waves (edges) per 256-thread block

  // ---- Layer 1: x[N x 128] @ W1[128 x 64] -> T ----
  gemm_wmma_k<128, 128, 64, false><<<dim3(Mtiles, 1, 1), dim3(32, 4, 1), 0, stream>>>(
      x, W1, nullptr, T, N);
  zero_f32<<<zblocks, 256, 0, stream>>>(agg, n64);
  scatter_edges<<<sblocks, 256, 0, stream>>>(T, src, dst, ew, agg, E);
  bias_relu_to_jk<<<zblocks, 256, 0, stream>>>(agg, b1, hjk, n64, 0);

  // ---- Layer 2: h1 (hjk cols 0..63, stride 192) @ W2[64 x 64] -> T ----
  gemm_wmma_k<64, 192, 64, false><<<dim3(Mtiles, 1, 1), dim3(32, 4, 1), 0, stream>>>(
      hjk, W2, nullptr, T, N);
  zero_f32<<<zblocks, 256, 0, stream>>>(agg, n64);
  scatter_edges<<<sblocks, 256, 0, stream>>>(T, src, dst, ew, agg, E);
  bias_relu_to_jk<<<zblocks, 256, 0, stream>>>(agg, b2, hjk, n64, 64);

  // ---- Layer 3: h2 (hjk cols 64..127) @ W3[64 x 64] -> T ----
  gemm_wmma_k<64, 192, 64, false><<<dim3(Mtiles, 1, 1), dim3(32, 4, 1), 0, stream>>>(
      hjk + 64, W3, nullptr, T, N);
  zero_f32<<<zblocks, 256, 0, stream>>>(agg, n64);
  scatter_edges<<<sblocks, 256, 0, stream>>>(T, src, dst, ew, agg, E);
  bias_relu_to_jk<<<zblocks, 256, 0, stream>>>(agg, b3, hjk, n64, 128);

  // ---- Final: hjk[N x 192] @ Wlin[192 x 40] + blin -> out[N x 40] ----
  gemm_wmma_k<192, 192, 40, true><<<dim3(Mtiles, 1, 1), dim3(32, 3, 1), 0, stream>>>(
      hjk, Wlin, blin, (float*)d_out, N);
}